// HeatmapLoss_57432302682511
// MI455X (gfx1250) — compile-verified
//
#include <hip/hip_runtime.h>
#include <math.h>

typedef __attribute__((ext_vector_type(2))) float v2f;
typedef __attribute__((ext_vector_type(8))) float v8f;

#define HMSZ 56
#define HW   (HMSZ * HMSZ)   // 3136
#define NF4  (HW / 4)        // 784  (56 divisible by 4 -> each float4 stays in one row)

// 1D windowed gaussian values: exp(-d^2 / (2*1.5^2)) for |d| <= 4, else 0
#define G1 0.8007374029168081f
#define G2 0.4111122905071874f
#define G3 0.1353352832366127f
#define G4 0.0285655007845504f
// (1 + 2*(G1+G2+G3+G4))^2 == KERNEL_NORM
#define SUM1D 3.7515009548903170
#define INV_NORM ((float)(1.0 / (SUM1D * SUM1D)))

// wing loss: x<2 ? 2*log1p(x/0.5) : x - C, with C = 2 - 2*ln(5)
#define NEG_C_WING 1.2188758248682006f

__device__ __forceinline__ float gwin(int d) {
    int a = d < 0 ? -d : d;
    float r = 0.0f;                 // |d| > 4 -> 0  (window clip)
    r = (a == 4) ? G4 : r;
    r = (a == 3) ? G3 : r;
    r = (a == 2) ? G2 : r;
    r = (a == 1) ? G1 : r;
    r = (a == 0) ? 1.0f : r;
    return r;
}

__device__ __forceinline__ float wing(float diff) {
    float x = fabsf(diff);
    return (x < 2.0f) ? 2.0f * log1pf(x * 2.0f) : (x + NEG_C_WING);
}

// ---------------------------------------------------------------------------
// Kernel 1: per-(b,n) streaming MSE vs analytic gaussian target.
// Wave reduction goes through V_WMMA_F32_16X16X4_F32 (A=partials, B=ones):
//   D[m][n] = partial[m] + partial[m+16]  -> 8 adds + shfl_xor(16) = wave sum.
// ---------------------------------------------------------------------------
__global__ __launch_bounds__(256) void k_hm_partial(
    const float* __restrict__ pred,      // (BN, 56, 56)
    const float2* __restrict__ tgt_lm,   // (BN) of (x,y) in image space
    float* __restrict__ partial)         // (BN) per-block sums
{
    const int bn  = blockIdx.x;
    const int tid = threadIdx.x;

    const float2 lm = tgt_lm[bn];
    const int xi = __float2int_rn(lm.x * 0.5f);   // scale = 56/112, rne == jnp.round
    const int yi = __float2int_rn(lm.y * 0.5f);

    const float4* p4 = (const float4*)(pred + (size_t)bn * HW);

    float acc = 0.0f;
    for (int i = tid; i < NF4; i += 256) {
        float4 v = p4[i];
        int y  = i / 14;               // 14 float4 per row of 56
        int x0 = (i - y * 14) * 4;
        float ty = gwin(y - yi) * INV_NORM;
        float t0 = ty * gwin(x0     - xi);
        float t1 = ty * gwin(x0 + 1 - xi);
        float t2 = ty * gwin(x0 + 2 - xi);
        float t3 = ty * gwin(x0 + 3 - xi);
        float d0 = v.x - t0, d1 = v.y - t1, d2 = v.z - t2, d3 = v.w - t3;
        acc = fmaf(d0, d0, acc);
        acc = fmaf(d1, d1, acc);
        acc = fmaf(d2, d2, acc);
        acc = fmaf(d3, d3, acc);
    }

    // ---- wave32 reduction via WMMA f32 16x16x4 (EXEC all-1s here) ----
    v2f a; a.x = acc;  a.y = 0.0f;     // lanes 0-15 -> (M=l,K=0); 16-31 -> (M=l-16,K=2)
    v2f b; b.x = 1.0f; b.y = 1.0f;     // B = ones
    v8f c = {0.f, 0.f, 0.f, 0.f, 0.f, 0.f, 0.f, 0.f};
    v8f d = __builtin_amdgcn_wmma_f32_16x16x4_f32(
        /*neg_a=*/false, a, /*neg_b=*/false, b,
        /*c_mod=*/(short)0, c, /*reuse_a=*/false, /*reuse_b=*/false);

    // lanes 0-15 hold rows 0-7 sums, lanes 16-31 hold rows 8-15 sums
    float s = d[0] + d[1] + d[2] + d[3] + d[4] + d[5] + d[6] + d[7];
    s += __shfl_xor(s, 16, 32);        // combine the two half-row groups

    __shared__ float red[8];
    const int lane = tid & 31, w = tid >> 5;
    if (lane == 0) red[w] = s;
    __syncthreads();
    if (tid == 0) {
        float tot = red[0];
        #pragma unroll
        for (int j = 1; j < 8; ++j) tot += red[j];
        partial[bn] = tot;             // deterministic: no float atomics
    }
}

__device__ __forceinline__ float block_reduce256(float v, float* sm) {
    const int tid = threadIdx.x;
    sm[tid] = v; __syncthreads();
    #pragma unroll
    for (int o = 128; o > 0; o >>= 1) {
        if (tid < o) sm[tid] += sm[tid + o];
        __syncthreads();
    }
    float r = sm[0]; __syncthreads();
    return r;
}

// ---------------------------------------------------------------------------
// Kernel 2 (single block): fold partials + all small loss terms, write 6 outs.
// ---------------------------------------------------------------------------
__global__ __launch_bounds__(256) void k_finalize(
    const float* __restrict__ partial, int nBN,
    const float* __restrict__ pred_lm,
    const float* __restrict__ pred_clm,
    const float* __restrict__ tgt_lm,
    const float* __restrict__ pred_gp, const float* __restrict__ tgt_gp,
    const float* __restrict__ pred_lp, const float* __restrict__ tgt_lp,
    int B, float* __restrict__ out)
{
    const int tid = threadIdx.x;
    const int nCoord = nBN * 2;
    const int nGp = B * 12;
    const int nLp = B * 40;

    float hsum = 0.0f;
    for (int i = tid; i < nBN; i += 256) hsum += partial[i];

    float csum = 0.0f, ksum = 0.0f;
    for (int i = tid; i < nCoord; i += 256) {
        float t = tgt_lm[i];
        csum += wing(pred_lm[i]  - t);
        ksum += wing(pred_clm[i] - t);
    }

    float gsum = 0.0f;
    for (int i = tid; i < nGp; i += 256) {
        float dd = pred_gp[i] - tgt_gp[i];
        gsum = fmaf(dd, dd, gsum);
    }

    float lsum = 0.0f;
    for (int i = tid; i < nLp; i += 256) lsum += fabsf(pred_lp[i] - tgt_lp[i]);

    __shared__ float sm[256];
    float H  = block_reduce256(hsum, sm);
    float Co = block_reduce256(csum, sm);
    float Ks = block_reduce256(ksum, sm);
    float Gp = block_reduce256(gsum, sm);
    float Lp = block_reduce256(lsum, sm);

    if (tid == 0) {
        float hm = H  / (float)((long long)nBN * HW);
        float co = Co / (float)nCoord;
        float cs = Ks / (float)nCoord;
        float gp = Gp / (float)nGp;
        float lp = Lp / (float)nLp;
        out[1] = hm; out[2] = co; out[3] = cs; out[4] = gp; out[5] = lp;
        out[0] = hm + 10.0f * co + 0.5f * cs + 0.01f * gp + 0.001f * lp;
    }
}

extern "C" void kernel_launch(void* const* d_in, const int* in_sizes, int n_in,
                              void* d_out, int out_size, void* d_ws, size_t ws_size,
                              hipStream_t stream) {
    const float* pred_hm  = (const float*)d_in[0];
    const float* pred_lm  = (const float*)d_in[1];
    const float* pred_clm = (const float*)d_in[2];
    const float* pred_gp  = (const float*)d_in[3];
    const float* pred_lp  = (const float*)d_in[4];
    const float* tgt_lm   = (const float*)d_in[5];
    const float* tgt_gp   = (const float*)d_in[6];
    const float* tgt_lp   = (const float*)d_in[7];

    const int B   = in_sizes[3] / 12;    // 256
    const int nBN = in_sizes[5] / 2;     // B*N = 17408

    float* partial = (float*)d_ws;       // nBN floats (~68 KB), fully rewritten each call

    k_hm_partial<<<nBN, 256, 0, stream>>>(pred_hm, (const float2*)tgt_lm, partial);
    k_finalize<<<1, 256, 0, stream>>>(partial, nBN, pred_lm, pred_clm, tgt_lm,
                                      pred_gp, tgt_gp, pred_lp, tgt_lp,
                                      B, (float*)d_out);
}